// PatchCore_51539607581
// MI455X (gfx1250) — compile-verified
//
#include <hip/hip_runtime.h>
#include <hip/hip_bf16.h>
#include <math.h>

// ---------------------------------------------------------------------------
// PatchCore pipeline for MI455X (gfx1250, wave32).
// Dominant cost: 12544x16384x128 f32 distance GEMM -> V_WMMA_F32_16X16X4_F32,
// with LDS-staged, double-buffered B tiles shared by all 8 waves of a block.
// ---------------------------------------------------------------------------

typedef __attribute__((ext_vector_type(2))) float v2f;
typedef __attribute__((ext_vector_type(8))) float v8f;

#define N_EMB   12544      // 16 * 28 * 28
#define M_BANK  16384
#define C_DIM   128
#define BATCH   16
#define HW      784        // 28*28
#define IMG     224
#define KSZ     33
#define PAD     16
#define SIGMA2X 32.0f      // 2 * sigma^2, sigma = 4

#define LDS_STRIDE 132     // 128 + 4 pad -> conflict-free ds_load_b64
#define M_TILES    (M_BANK / 16)   // 1024

// ---------------------------------------------------------------------------
// Kernel 1: memory bank squared norms
// ---------------------------------------------------------------------------
__global__ __launch_bounds__(256) void mnorm_kernel(const float* __restrict__ bank,
                                                    float* __restrict__ mnorm) {
    const int m = blockIdx.x * 256 + threadIdx.x;
    if (m >= M_BANK) return;
    const float4* row = (const float4*)(bank + (size_t)m * C_DIM);
    float s = 0.f;
#pragma unroll 8
    for (int k = 0; k < C_DIM / 4; ++k) {
        float4 v = row[k];
        s += v.x * v.x + v.y * v.y + v.z * v.z + v.w * v.w;
    }
    mnorm[m] = s;
}

// ---------------------------------------------------------------------------
// Kernel 2: fused WMMA GEMM + per-row 1-NN argmin.
// Block = 128 embedding rows (8 waves x 16 rows). All waves share each
// 16x128 bank tile, cooperatively loaded into a double-buffered LDS tile.
// ---------------------------------------------------------------------------
__global__ __launch_bounds__(256) void gemm_argmin_kernel(
    const float* __restrict__ emb, const float* __restrict__ bank,
    const float* __restrict__ mnorm,
    float* __restrict__ patch_scores, int* __restrict__ locations) {

    const int tid  = threadIdx.x;
    const int lane = tid & 31;
    const int wave = tid >> 5;                   // 0..7
    const int half = lane >> 4;                  // 0/1 (K sub-pair)
    const int l16  = lane & 15;

    __shared__ float ldsB[2][16 * LDS_STRIDE];   // double-buffered 16x128 tile

    // ---- per-wave A fragments: 16 rows x 128 K held in VGPRs ----
    // A 16x4 f32 layout: lane L holds row L%16, K = 4*kk + 2*(L/16) + {0,1}
    const int rowbase = blockIdx.x * 128 + wave * 16;
    const int arow    = rowbase + l16;
    const float* abase = emb + (size_t)arow * C_DIM + half * 2;
    v2f afrag[32];
    float enorm = 0.f;
#pragma unroll
    for (int kk = 0; kk < 32; ++kk) {
        v2f a = *(const v2f*)(abase + kk * 4);
        afrag[kk] = a;
        enorm += a.x * a.x + a.y * a.y;
    }
    enorm += __shfl_xor(enorm, 16, 32);          // full ||e_row||^2 at lane row

    // ---- cooperative B-tile staging: 256 thr x 8 floats = 16x128 tile ----
    const int crow = tid >> 4;                   // 0..15  (bank row in tile)
    const int ccol = (tid & 15) * 8;             // 0..120 (col, 8 floats)

    {   // preload tile 0
        const float* src = bank + (size_t)crow * C_DIM + ccol;
        float4 r0 = *(const float4*)(src);
        float4 r1 = *(const float4*)(src + 4);
        float* d = &ldsB[0][crow * LDS_STRIDE + ccol];
        *(float4*)(d)     = r0;
        *(float4*)(d + 4) = r1;
    }
    __syncthreads();

    float best[8];
    int   bidx[8];
#pragma unroll
    for (int v = 0; v < 8; ++v) { best[v] = 3.4e38f; bidx[v] = 0; }

    for (int mt = 0; mt < M_TILES; ++mt) {
        const int p  = mt & 1;
        const int m0 = mt * 16;

        // issue global loads for next tile early (overlap with WMMA compute)
        float4 r0, r1;
        const bool have_next = (mt + 1 < M_TILES);
        if (have_next) {
            const float* src = bank + (size_t)(m0 + 16 + crow) * C_DIM + ccol;
            r0 = *(const float4*)(src);
            r1 = *(const float4*)(src + 4);
            if (mt + 2 < M_TILES)
                __builtin_prefetch(bank + (size_t)(m0 + 32 + crow) * C_DIM + ccol, 0, 1);
        }

        // 16x16 tile: 32 chained WMMAs, B fragments from LDS in groups of 8
        v8f acc = {};
        const float* bb = &ldsB[p][l16 * LDS_STRIDE + half * 2];
#pragma unroll
        for (int g = 0; g < 4; ++g) {
            v2f bf[8];
#pragma unroll
            for (int j = 0; j < 8; ++j)
                bf[j] = *(const v2f*)(bb + (g * 8 + j) * 4);
#pragma unroll
            for (int j = 0; j < 8; ++j)
                acc = __builtin_amdgcn_wmma_f32_16x16x4_f32(
                    /*neg_a=*/false, afrag[g * 8 + j], /*neg_b=*/false, bf[j],
                    /*c_mod=*/(short)0, acc, /*reuse_a=*/false, /*reuse_b=*/false);
        }

        // argmin update on (||m||^2 - 2 e.m)
        const float mn = mnorm[m0 + l16];
#pragma unroll
        for (int v = 0; v < 8; ++v) {
            // C element: emb row (v + 8*half), bank col (m0 + l16)
            const float cand = mn - 2.0f * acc[v];
            if (cand < best[v]) { best[v] = cand; bidx[v] = m0 + l16; }
        }

        if (have_next) {
            float* d = &ldsB[1 - p][crow * LDS_STRIDE + ccol];
            *(float4*)(d)     = r0;
            *(float4*)(d + 4) = r1;
        }
        __syncthreads();
    }

    // ---- in-wave reduction across the 16 bank columns, then write ----
#pragma unroll
    for (int v = 0; v < 8; ++v) {
        float bv = best[v];
        int   bi = bidx[v];
#pragma unroll
        for (int off = 1; off < 16; off <<= 1) {
            float ov = __shfl_xor(bv, off, 32);
            int   oi = __shfl_xor(bi, off, 32);
            if (ov < bv) { bv = ov; bi = oi; }
        }
        const float en = __shfl(enorm, v + 8 * half, 32);  // row norm for this slot
        if (l16 == 0) {
            const int r = rowbase + v + 8 * half;
            patch_scores[r] = sqrtf(fmaxf(en + bv, 1e-12f));
            locations[r]    = bi;
        }
    }
}

// ---------------------------------------------------------------------------
// Kernel 3: per-batch argmax of patch scores
// ---------------------------------------------------------------------------
__global__ __launch_bounds__(256) void batch_max_kernel(
    const float* __restrict__ ps, const int* __restrict__ loc,
    float* __restrict__ score, int* __restrict__ nn_index,
    int* __restrict__ max_patch) {
    __shared__ float sv[256];
    __shared__ int   si[256];
    const int b = blockIdx.x;
    const int tid = threadIdx.x;
    float best = -3.4e38f; int bi = 0;
    for (int i = tid; i < HW; i += 256) {
        float v = ps[b * HW + i];
        if (v > best) { best = v; bi = i; }
    }
    sv[tid] = best; si[tid] = bi;
    __syncthreads();
    for (int s = 128; s > 0; s >>= 1) {
        if (tid < s && sv[tid + s] > sv[tid]) { sv[tid] = sv[tid + s]; si[tid] = si[tid + s]; }
        __syncthreads();
    }
    if (tid == 0) {
        score[b]     = sv[0];
        max_patch[b] = si[0];
        nn_index[b]  = loc[b * HW + si[0]];
    }
}

// ---------------------------------------------------------------------------
// Kernel 4: d_nn[b,m] = || bank[nn_index[b]] - bank[m] ||
// ---------------------------------------------------------------------------
__global__ __launch_bounds__(256) void dnn_kernel(
    const float* __restrict__ bank, const float* __restrict__ mnorm,
    const int* __restrict__ nn_index, float* __restrict__ dnn) {
    const int idx = blockIdx.x * 256 + threadIdx.x;    // 16*16384 threads
    const int b = idx >> 14;
    const int m = idx & (M_BANK - 1);
    const int nn = nn_index[b];
    const float4* a = (const float4*)(bank + (size_t)nn * C_DIM);
    const float4* c = (const float4*)(bank + (size_t)m * C_DIM);
    float dot = 0.f;
#pragma unroll 8
    for (int k = 0; k < C_DIM / 4; ++k) {
        float4 va = a[k], vc = c[k];
        dot += va.x * vc.x + va.y * vc.y + va.z * vc.z + va.w * vc.w;
    }
    const float d2 = mnorm[nn] + mnorm[m] - 2.0f * dot;
    dnn[idx] = sqrtf(fmaxf(d2, 1e-12f));
}

// ---------------------------------------------------------------------------
// Kernel 5: 9 smallest entries per d_nn row (iterative block argmin;
//           destroys dnn, which is regenerated every launch)
// ---------------------------------------------------------------------------
__global__ __launch_bounds__(256) void topk_kernel(float* __restrict__ dnn,
                                                   int* __restrict__ support) {
    __shared__ float sv[256];
    __shared__ int   si[256];
    const int b = blockIdx.x;
    const int tid = threadIdx.x;
    for (int k = 0; k < 9; ++k) {
        float best = 3.4e38f; int bi = 0;
        for (int m = tid; m < M_BANK; m += 256) {
            float v = dnn[b * M_BANK + m];
            if (v < best) { best = v; bi = m; }
        }
        sv[tid] = best; si[tid] = bi;
        __syncthreads();
        for (int s = 128; s > 0; s >>= 1) {
            if (tid < s && sv[tid + s] < sv[tid]) { sv[tid] = sv[tid + s]; si[tid] = si[tid + s]; }
            __syncthreads();
        }
        if (tid == 0) {
            support[b * 9 + k] = si[0];
            dnn[b * M_BANK + si[0]] = 3.4e38f;
        }
        __syncthreads();
    }
}

// ---------------------------------------------------------------------------
// Kernel 6: softmax re-weighted prediction score
// ---------------------------------------------------------------------------
__global__ __launch_bounds__(32) void score_kernel(
    const float* __restrict__ emb, const float* __restrict__ bank,
    const int* __restrict__ support, const int* __restrict__ max_patch,
    const float* __restrict__ score, float* __restrict__ out_pred) {
    const int b = threadIdx.x;
    if (b >= BATCH) return;
    const float* f = emb + ((size_t)b * HW + max_patch[b]) * C_DIM;
    float d[9];
#pragma unroll
    for (int k = 0; k < 9; ++k) {
        const float* s = bank + (size_t)support[b * 9 + k] * C_DIM;
        float acc = 0.f;
        for (int c = 0; c < C_DIM; ++c) {
            float df = f[c] - s[c];
            acc += df * df;
        }
        d[k] = sqrtf(fmaxf(acc, 1e-12f));
    }
    float mx = d[0];
#pragma unroll
    for (int k = 1; k < 9; ++k) mx = fmaxf(mx, d[k]);
    float sum = 0.f;
#pragma unroll
    for (int k = 0; k < 9; ++k) sum += expf(d[k] - mx);
    const float w = 1.0f - expf(d[0] - mx) / sum;
    out_pred[b] = w * score[b];
}

// ---------------------------------------------------------------------------
// Kernels 7/8: nearest-upsample (28->224) fused with separable 33-tap
// reflect-padded Gaussian blur
// ---------------------------------------------------------------------------
__device__ __forceinline__ int refl(int p) {
    if (p < 0) return -p;
    if (p > IMG - 1) return 2 * (IMG - 1) - p;
    return p;
}

__global__ __launch_bounds__(256) void hblur_kernel(const float* __restrict__ ps,
                                                    float* __restrict__ tmp) {
    __shared__ float g[KSZ];
    __shared__ float gsum;
    if (threadIdx.x < KSZ) {
        float x = (float)((int)threadIdx.x - PAD);
        g[threadIdx.x] = expf(-(x * x) / SIGMA2X);
    }
    __syncthreads();
    if (threadIdx.x == 0) {
        float s = 0.f;
        for (int i = 0; i < KSZ; ++i) s += g[i];
        gsum = s;
    }
    __syncthreads();

    const int gid = blockIdx.x * 256 + threadIdx.x;   // 16*224*224
    const int b = gid / (IMG * IMG);
    const int rem = gid - b * (IMG * IMG);
    const int y = rem / IMG;
    const int x = rem - y * IMG;
    const int prow = b * HW + (y >> 3) * 28;
    float acc = 0.f;
#pragma unroll
    for (int t = 0; t < KSZ; ++t) {
        const int xs = refl(x + t - PAD);
        acc += g[t] * ps[prow + (xs >> 3)];
    }
    tmp[gid] = acc / gsum;
}

__global__ __launch_bounds__(256) void vblur_kernel(const float* __restrict__ tmp,
                                                    float* __restrict__ out) {
    __shared__ float g[KSZ];
    __shared__ float gsum;
    if (threadIdx.x < KSZ) {
        float x = (float)((int)threadIdx.x - PAD);
        g[threadIdx.x] = expf(-(x * x) / SIGMA2X);
    }
    __syncthreads();
    if (threadIdx.x == 0) {
        float s = 0.f;
        for (int i = 0; i < KSZ; ++i) s += g[i];
        gsum = s;
    }
    __syncthreads();

    const int gid = blockIdx.x * 256 + threadIdx.x;
    const int b = gid / (IMG * IMG);
    const int rem = gid - b * (IMG * IMG);
    const int y = rem / IMG;
    const int x = rem - y * IMG;
    float acc = 0.f;
#pragma unroll
    for (int t = 0; t < KSZ; ++t) {
        const int ys = refl(y + t - PAD);
        acc += g[t] * tmp[b * (IMG * IMG) + ys * IMG + x];
    }
    out[gid] = acc / gsum;
}

// ---------------------------------------------------------------------------
// Host-side launcher
// ---------------------------------------------------------------------------
extern "C" void kernel_launch(void* const* d_in, const int* in_sizes, int n_in,
                              void* d_out, int out_size, void* d_ws, size_t ws_size,
                              hipStream_t stream) {
    const float* emb  = (const float*)d_in[0];   // [12544, 128]
    const float* bank = (const float*)d_in[1];   // [16384, 128]

    // workspace layout (16-byte aligned slices)
    char* base = (char*)d_ws;
    size_t off = 0;
    auto alloc = [&](size_t bytes) {
        void* p = base + off;
        off += (bytes + 15) & ~(size_t)15;
        return p;
    };
    float* ps        = (float*)alloc(N_EMB * sizeof(float));
    int*   loc       = (int*)  alloc(N_EMB * sizeof(int));
    float* mnorm     = (float*)alloc(M_BANK * sizeof(float));
    float* score     = (float*)alloc(BATCH * sizeof(float));
    int*   nn_index  = (int*)  alloc(BATCH * sizeof(int));
    int*   max_patch = (int*)  alloc(BATCH * sizeof(int));
    float* dnn       = (float*)alloc((size_t)BATCH * M_BANK * sizeof(float));
    int*   support   = (int*)  alloc(BATCH * 9 * sizeof(int));
    float* tmp       = (float*)alloc((size_t)BATCH * IMG * IMG * sizeof(float));

    float* out_map  = (float*)d_out;                       // [16,1,224,224]
    float* out_pred = (float*)d_out + BATCH * IMG * IMG;   // [16]

    mnorm_kernel<<<M_BANK / 256, 256, 0, stream>>>(bank, mnorm);
    gemm_argmin_kernel<<<N_EMB / 128, 256, 0, stream>>>(emb, bank, mnorm, ps, loc);
    batch_max_kernel<<<BATCH, 256, 0, stream>>>(ps, loc, score, nn_index, max_patch);
    dnn_kernel<<<(BATCH * M_BANK) / 256, 256, 0, stream>>>(bank, mnorm, nn_index, dnn);
    topk_kernel<<<BATCH, 256, 0, stream>>>(dnn, support);
    score_kernel<<<1, 32, 0, stream>>>(emb, bank, support, max_patch, score, out_pred);
    hblur_kernel<<<(BATCH * IMG * IMG) / 256, 256, 0, stream>>>(ps, tmp);
    vblur_kernel<<<(BATCH * IMG * IMG) / 256, 256, 0, stream>>>(tmp, out_map);
}